// TorchFFAmoeba_7559142441138
// MI455X (gfx1250) — compile-verified
//
#include <hip/hip_runtime.h>
#include <math.h>

#define THREADS 256

__global__ void zero_out_kernel(float* __restrict__ out, int n) {
    int i = blockIdx.x * blockDim.x + threadIdx.x;
    if (i < n) out[i] = 0.0f;
}

__global__ __launch_bounds__(THREADS)
void amoeba_pair_kernel(const float* __restrict__ coords,
                        const float* __restrict__ box,
                        const int*   __restrict__ pairs,
                        const float* __restrict__ q,
                        const float* __restrict__ p,
                        const float* __restrict__ t,
                        const float* __restrict__ polarity,
                        float* __restrict__ out,   // [0]=energy, [1..3N]=field
                        int P)
{
    constexpr float CUTOFF = 3.0f;
    constexpr float ALPHA  = 3.0f;
    constexpr float THOLE  = 0.39f;
    constexpr float PREF   = 138.935456f;
    constexpr float CSQ    = 1.12837916709551257390f; // 2/sqrt(pi)

    __shared__ int   s_pairs[2 * THREADS];
    __shared__ float s_ewave[THREADS / 32];

    const int tid = threadIdx.x;
    const int idx = blockIdx.x * blockDim.x + tid;
    const bool valid = (idx < P);

    // ---- CDNA5 async-to-LDS: stage this lane's pair indices (8B) into LDS,
    // overlapping the copy with the uniform box-inverse computation below.
    if (valid) {
        const int* gsrc = pairs + 2 * (size_t)idx;
        unsigned ldsoff = (unsigned)(uintptr_t)(&s_pairs[2 * tid]); // AS3 offset in low 32 bits
        asm volatile("global_load_async_to_lds_b64 %0, %1, off"
                     :: "v"(ldsoff), "v"(gsrc) : "memory");
    }

    // ---- Uniform work while async copy is in flight: 3x3 box inverse.
    const float b00 = box[0], b01 = box[1], b02 = box[2];
    const float b10 = box[3], b11 = box[4], b12 = box[5];
    const float b20 = box[6], b21 = box[7], b22 = box[8];
    const float c00 = b11 * b22 - b12 * b21;
    const float c01 = b12 * b20 - b10 * b22;
    const float c02 = b10 * b21 - b11 * b20;
    const float invdet = 1.0f / (b00 * c00 + b01 * c01 + b02 * c02);
    const float i00 = c00 * invdet;
    const float i01 = (b02 * b21 - b01 * b22) * invdet;
    const float i02 = (b01 * b12 - b02 * b11) * invdet;
    const float i10 = c01 * invdet;
    const float i11 = (b00 * b22 - b02 * b20) * invdet;
    const float i12 = (b02 * b10 - b00 * b12) * invdet;
    const float i20 = c02 * invdet;
    const float i21 = (b01 * b20 - b00 * b21) * invdet;
    const float i22 = (b00 * b11 - b01 * b10) * invdet;

    // ---- Wait for the async engine, then read pair from LDS.
    asm volatile("s_wait_asynccnt 0" ::: "memory");

    float esum = 0.0f;

    if (valid) {
        const int ii = s_pairs[2 * tid + 0];
        const int jj = s_pairs[2 * tid + 1];

        // Prefetch the widest per-atom gathers (quadrupole rows, 36B each) so the
        // L2 fetch overlaps the minimum-image + damping math (global_prefetch_b8).
        const float* ti = t + 9 * (size_t)ii;
        const float* tj = t + 9 * (size_t)jj;
        __builtin_prefetch(ti, 0, 3);
        __builtin_prefetch(tj, 0, 3);

        // Minimum-image displacement R = x_j - x_i
        float dx = coords[3 * jj + 0] - coords[3 * ii + 0];
        float dy = coords[3 * jj + 1] - coords[3 * ii + 1];
        float dz = coords[3 * jj + 2] - coords[3 * ii + 2];
        float s0 = rintf(dx * i00 + dy * i10 + dz * i20);
        float s1 = rintf(dx * i01 + dy * i11 + dz * i21);
        float s2 = rintf(dx * i02 + dy * i12 + dz * i22);
        dx -= s0 * b00 + s1 * b10 + s2 * b20;
        dy -= s0 * b01 + s1 * b11 + s2 * b21;
        dz -= s0 * b02 + s1 * b12 + s2 * b22;

        const float r2 = dx * dx + dy * dy + dz * dz;
        const float r  = sqrtf(r2);

        if (r <= CUTOFF) {
            const float rinv  = 1.0f / r;
            const float rinv2 = rinv * rinv;
            const float r3inv = rinv * rinv2;
            const float r5inv = r3inv * rinv2;
            const float r7inv = r5inv * rinv2;
            const float r9inv = r7inv * rinv2;

            // --- erfc damping (permanent)
            const float u  = ALPHA * r;
            const float u2 = u * u;
            const float eg = expf(-u2);
            const float d1 = erfcf(u);
            const float ue = CSQ * u * eg;
            const float u4 = u2 * u2;
            const float d3 = d1 + ue;
            const float d5 = d3 + ue * (2.0f / 3.0f) * u2;
            const float d7 = d5 + ue * (4.0f / 15.0f) * u4;
            const float d9 = d7 + ue * (8.0f / 105.0f) * u4 * u2;

            const float rr1 = d1 * rinv;
            const float rr3 = d3 * r3inv;
            const float rr5 = d5 * r5inv;
            const float rr7 = d7 * r7inv;
            const float rr9 = d9 * r9inv;

            // --- Thole damping (polarization); (pol_i pol_j)^(-1/2) == factor^3
            const float x  = THOLE * (r2 * r) / sqrtf(polarity[ii] * polarity[jj]);
            const float ex = expf(-x);
            const float x2 = x * x;
            const float rr3p = (d3 - ex) * r3inv;
            const float rr5p = (d5 - (1.0f + x) * ex) * r5inv;
            const float rr7p = (d7 - (1.0f + x + 0.6f * x2) * ex) * r7inv;

            // --- Gather multipoles
            const float qi = q[ii], qj = q[jj];
            const float pix = p[3 * ii + 0], piy = p[3 * ii + 1], piz = p[3 * ii + 2];
            const float pjx = p[3 * jj + 0], pjy = p[3 * jj + 1], pjz = p[3 * jj + 2];
            const float ti00 = ti[0], ti01 = ti[1], ti02 = ti[2], ti11 = ti[4], ti12 = ti[5], ti22 = ti[8];
            const float tj00 = tj[0], tj01 = tj[1], tj02 = tj[2], tj11 = tj[4], tj12 = tj[5], tj22 = tj[8];

            // --- Invariants (t is symmetric traceless)
            const float pRi = pix * dx + piy * dy + piz * dz;
            const float pRj = pjx * dx + pjy * dy + pjz * dz;
            const float pp  = pix * pjx + piy * pjy + piz * pjz;

            const float QiRx = ti00 * dx + ti01 * dy + ti02 * dz;
            const float QiRy = ti01 * dx + ti11 * dy + ti12 * dz;
            const float QiRz = ti02 * dx + ti12 * dy + ti22 * dz;
            const float QjRx = tj00 * dx + tj01 * dy + tj02 * dz;
            const float QjRy = tj01 * dx + tj11 * dy + tj12 * dz;
            const float QjRz = tj02 * dx + tj12 * dy + tj22 * dz;

            const float RQiR = dx * QiRx + dy * QiRy + dz * QiRz;
            const float RQjR = dx * QjRx + dy * QjRy + dz * QjRz;
            const float pjQiR = pjx * QiRx + pjy * QiRy + pjz * QiRz;
            const float piQjR = pix * QjRx + piy * QjRy + piz * QjRz;
            const float QQdot = QiRx * QjRx + QiRy * QjRy + QiRz * QjRz;
            const float QiQj  = ti00 * tj00 + ti11 * tj11 + ti22 * tj22
                              + 2.0f * (ti01 * tj01 + ti02 * tj02 + ti12 * tj12);

            // --- Permanent multipole energy m_j^T T_perm m_i (closed form)
            const float E =
                  rr1 * qi * qj
                + rr3 * (qj * pRi - qi * pRj + pp)
                + rr5 * (3.0f * (qj * RQiR + qi * RQjR) - 3.0f * pRi * pRj
                         + 6.0f * (pjQiR - piQjR) + 6.0f * QiQj)
                + rr7 * (15.0f * (pRi * RQjR - pRj * RQiR) - 60.0f * QQdot)
                + rr9 * (105.0f * RQiR * RQjR);
            esum = PREF * E;

            // --- Polarization fields (dipole rows/cols of T_pol)
            // e_ij (deposited at jj): field of multipole i
            const float ci = -qi * rr3p - 3.0f * pRi * rr5p - 15.0f * RQiR * rr7p;
            const float eijx = ci * dx + pix * rr3p + 6.0f * QiRx * rr5p;
            const float eijy = ci * dy + piy * rr3p + 6.0f * QiRy * rr5p;
            const float eijz = ci * dz + piz * rr3p + 6.0f * QiRz * rr5p;
            // e_ji (deposited at ii): field of multipole j
            const float cj = qj * rr3p - 3.0f * pRj * rr5p + 15.0f * RQjR * rr7p;
            const float ejix = cj * dx + pjx * rr3p - 6.0f * QjRx * rr5p;
            const float ejiy = cj * dy + pjy * rr3p - 6.0f * QjRy * rr5p;
            const float ejiz = cj * dz + pjz * rr3p - 6.0f * QjRz * rr5p;

            float* fj = out + 1 + 3 * (size_t)jj;
            atomicAdd(&fj[0], -PREF * eijx);
            atomicAdd(&fj[1], -PREF * eijy);
            atomicAdd(&fj[2], -PREF * eijz);
            float* fi = out + 1 + 3 * (size_t)ii;
            atomicAdd(&fi[0], -PREF * ejix);
            atomicAdd(&fi[1], -PREF * ejiy);
            atomicAdd(&fi[2], -PREF * ejiz);
        }
    }

    // ---- Two-level energy reduction: wave32 shuffle, then LDS, one atomic/block.
    #pragma unroll
    for (int off = 16; off > 0; off >>= 1)
        esum += __shfl_xor(esum, off, 32);
    if ((tid & 31) == 0)
        s_ewave[tid >> 5] = esum;
    __syncthreads();
    if (tid < (THREADS / 32)) {
        float v = s_ewave[tid];
        #pragma unroll
        for (int off = (THREADS / 64); off > 0; off >>= 1)
            v += __shfl_xor(v, off, 32);
        if (tid == 0)
            atomicAdd(&out[0], v);
    }
}

extern "C" void kernel_launch(void* const* d_in, const int* in_sizes, int n_in,
                              void* d_out, int out_size, void* d_ws, size_t ws_size,
                              hipStream_t stream) {
    const float* coords   = (const float*)d_in[0];
    const float* box      = (const float*)d_in[1];
    const int*   pairs    = (const int*)  d_in[2];
    const float* q        = (const float*)d_in[3];
    const float* p        = (const float*)d_in[4];
    const float* t        = (const float*)d_in[5];
    const float* polarity = (const float*)d_in[6];
    float* out = (float*)d_out;

    const int P = in_sizes[2] / 2;

    zero_out_kernel<<<(out_size + THREADS - 1) / THREADS, THREADS, 0, stream>>>(out, out_size);
    amoeba_pair_kernel<<<(P + THREADS - 1) / THREADS, THREADS, 0, stream>>>(
        coords, box, pairs, q, p, t, polarity, out, P);
}